// SegHead_65008624992826
// MI455X (gfx1250) — compile-verified
//
#include <hip/hip_runtime.h>

// SegHead: segment-max over (B*K, C) + transpose copy (B,C,N) -> (B*N, C).
// Bandwidth problem (~256 MiB min traffic -> ~11us @ 23.3 TB/s). CDNA5 path:
// async global->LDS staging and async LDS->global transposed stores (ASYNCcnt),
// ds_max_u32 segment accumulation in a 64KB LDS accumulator (320KB WGP LDS),
// global_prefetch_b8 stream-ahead, NT temporal hints (working set > 192MB L2).

namespace {
constexpr int B = 8, C = 256, N = 16384, K = 128;
constexpr int CHUNK = 256;          // points (n) per block
constexpr int NBLK  = N / CHUNK;    // 64 n-chunks
constexpr int CPG   = 128;          // channels per block (C split in 2)
constexpr int CSUB  = 32;           // channels staged per LDS tile pass
constexpr int TPAD  = 36;           // tile row stride (dwords): 144B, 16B-aligned
constexpr int SEG   = B * K * C;    // 262144 f32 = output-0 element count
}

// Monotone order-preserving f32 <-> u32 map: max over uints == max over floats.
__device__ __forceinline__ unsigned enc_f32(float f) {
  unsigned u = __float_as_uint(f);
  return (u & 0x80000000u) ? ~u : (u | 0x80000000u);
}
__device__ __forceinline__ float dec_f32(unsigned u) {
  unsigned b = (u & 0x80000000u) ? (u ^ 0x80000000u) : ~u;
  return __uint_as_float(b);
}

// --- gfx1250 async global<->LDS ops (ASYNCcnt-tracked), GV addressing ------
__device__ __forceinline__ void async_load_b32(unsigned lds_off, const void* g) {
  asm volatile("global_load_async_to_lds_b32 %0, %1, off th:TH_LOAD_NT"
               :: "v"(lds_off), "v"((unsigned long long)(size_t)g)
               : "memory");
}
__device__ __forceinline__ void async_store_b32(void* g, unsigned lds_off) {
  asm volatile("global_store_async_from_lds_b32 %0, %1, off th:TH_STORE_NT"
               :: "v"((unsigned long long)(size_t)g), "v"(lds_off)
               : "memory");
}
__device__ __forceinline__ void wait_async0() {
#if __has_builtin(__builtin_amdgcn_s_wait_asynccnt)
  __builtin_amdgcn_s_wait_asynccnt(0);
#else
  asm volatile("s_wait_asynccnt 0x0" ::: "memory");
#endif
}

__global__ __launch_bounds__(256) void seg_zero_acc(unsigned* __restrict__ acc) {
  acc[(size_t)blockIdx.x * 256 + threadIdx.x] = 0u;
}

// One block = (batch b, 256-point n-chunk, one 128-channel half of C).
// 1024 blocks x 256 threads (8 wave32 waves). LDS: 64KB accumulator + 36KB
// tile (~100KB -> ~3 workgroups per 320KB WGP).
__global__ __launch_bounds__(256) void seg_max_fused(
    const float* __restrict__ feat,      // (B, C, N)
    const int*   __restrict__ cid,       // (B, N), -1..K-1
    unsigned*    __restrict__ gacc,      // (B*K, C) encoded-u32 accumulator
    float*       __restrict__ pf_out) {  // (B*N, C)
  __shared__ unsigned acc[K * CPG];            // 64 KB
  __shared__ float    tile[CHUNK][TPAD];       // 36 KB, tile[n][cc]

  const int tid  = threadIdx.x;
  const int lane = tid & 31;
  const int wrp  = tid >> 5;

  for (int i = tid; i < K * CPG; i += 256) acc[i] = 0u;

  const int b  = blockIdx.x / (NBLK * 2);
  const int r  = blockIdx.x % (NBLK * 2);
  const int n0 = (r >> 1) * CHUNK;
  const int c0 = (r & 1) * CPG;

  const int myk    = cid[b * N + n0 + tid];        // this thread's point label
  const int mybase = (myk >= 0) ? myk * CPG : -1;  // -1 => dump segment, drop

  const float* fb = feat + (size_t)b * C * N + n0; // row c lives at fb + c*N
  // Low 32 bits of a generic pointer into LDS == workgroup LDS byte offset.
  const unsigned tile_lds = (unsigned)(size_t)&tile[0][0];
  const unsigned row_lds  = tile_lds + (unsigned)tid * (TPAD * 4u);
  __syncthreads();

  for (int cs = 0; cs < CPG; cs += CSUB) {
    // Phase 1: async global->LDS stage, tile[n][cc] = feat[b][c0+cs+cc][n0+n].
    // Per wave: 32 lanes x 4B contiguous = 128B per async op; no VGPR data.
    #pragma unroll 8
    for (int cc = 0; cc < CSUB; ++cc) {
      const size_t c = (size_t)(c0 + cs + cc);
      async_load_b32(row_lds + 4u * (unsigned)cc, fb + c * N + tid);
      if (c + 8 < C)  // stream-ahead: global_prefetch_b8
        __builtin_prefetch(fb + (c + 8) * N + tid, 0, 1);
    }
    wait_async0();
    __syncthreads();

    // Phase 2: ds_load_b128 readback of this thread's point row (16B-aligned,
    // lane-consecutive -> conflict-free) + native ds_max_u32 accumulation.
    if (mybase >= 0) {
      #pragma unroll
      for (int q = 0; q < CSUB / 4; ++q) {
        const float4 v = *(const float4*)&tile[tid][q * 4];
        atomicMax(&acc[mybase + cs + q * 4 + 0], enc_f32(v.x));
        atomicMax(&acc[mybase + cs + q * 4 + 1], enc_f32(v.y));
        atomicMax(&acc[mybase + cs + q * 4 + 2], enc_f32(v.z));
        atomicMax(&acc[mybase + cs + q * 4 + 3], enc_f32(v.w));
      }
    }

    // Phase 3: transposed async LDS->global stores: pf_out[(b*N+n)*C + c].
    // Each wave stores 32 consecutive channels = 128B contiguous per row;
    // LDS source addresses are lane-consecutive (stride-1 in a tile row).
    float* ob = pf_out + ((size_t)b * N + n0) * C + c0 + cs;
    #pragma unroll 8
    for (int rr = 0; rr < CHUNK / 8; ++rr) {
      const int n = wrp + rr * 8;
      async_store_b32(ob + (size_t)n * C + lane,
                      tile_lds + (unsigned)(n * TPAD + lane) * 4u);
    }
    wait_async0();      // stores must finish reading tile before reuse
    __syncthreads();
  }

  // Merge block-local maxima into the global accumulator (encoded u32,
  // global_atomic_max_u32, no return). Skip still-empty slots.
  unsigned* gw = gacc + (size_t)b * K * C + c0;
  for (int i = tid; i < K * CPG; i += 256) {
    const unsigned v = acc[i];
    if (v) atomicMax(&gw[(size_t)(i >> 7) * C + (i & (CPG - 1))], v);
  }
}

// Decode in place: encoded 0 == empty segment -> 0.0f (matches the
// reference's where(isfinite(segment_max), x, 0)).
__global__ __launch_bounds__(256) void seg_finalize(unsigned* __restrict__ io) {
  const size_t i = (size_t)blockIdx.x * 256 + threadIdx.x;
  const unsigned v = io[i];
  ((float*)io)[i] = v ? dec_f32(v) : 0.0f;
}

extern "C" void kernel_launch(void* const* d_in, const int* in_sizes, int n_in,
                              void* d_out, int out_size, void* d_ws, size_t ws_size,
                              hipStream_t stream) {
  (void)in_sizes; (void)n_in; (void)d_ws; (void)ws_size; (void)out_size;
  const float* feat = (const float*)d_in[0];   // (B, C, N) f32
  const int*   cid  = (const int*)d_in[1];     // (B, N) i32
  float* out = (float*)d_out;                  // [SEG seg_feats | B*N*C pf_flat]

  // Segment accumulator lives in-place in d_out's first SEG words as encoded
  // u32; zeroed every call (harness does not re-poison between replays).
  unsigned* acc = (unsigned*)d_out;

  seg_zero_acc<<<SEG / 256, 256, 0, stream>>>(acc);
  seg_max_fused<<<B * NBLK * 2, 256, 0, stream>>>(feat, cid, acc, out + SEG);
  seg_finalize<<<SEG / 256, 256, 0, stream>>>(acc);
}